// ILif3f_3315714752610
// MI455X (gfx1250) — compile-verified
//
#include <hip/hip_runtime.h>
#include <math.h>

#define N_IN 4096
#define N_REC 4096
#define ROW_SPLIT 8
#define ROWS_PER_CHUNK (N_IN / ROW_SPLIT)   // 512
#define COL_GROUPS (N_REC / 16)             // 256

typedef __attribute__((ext_vector_type(2))) float v2f;
typedef __attribute__((ext_vector_type(8))) float v8f;

// ---------------------------------------------------------------------------
// Kernel 1: GEMV  isc[j] = sum_i inputs[i] * w_in[i][j]
// via V_WMMA_F32_16X16X4_F32. One wave owns a (16-col, 512-row) tile.
// 2048 waves total = 256 blocks x 8 waves.
// ---------------------------------------------------------------------------
__global__ __launch_bounds__(256) void gemv_wmma_kernel(
    const float* __restrict__ inputs,
    const float* __restrict__ w_in,
    float* __restrict__ partials /* [ROW_SPLIT][N_REC] */) {
  const int wave = threadIdx.x >> 5;
  const int lane = threadIdx.x & 31;
  const int tile = blockIdx.x * 8 + wave;        // 0..2047
  const int colGroup = tile & (COL_GROUPS - 1);  // 0..255 (fastest -> adjacent waves read adjacent cols)
  const int rowChunk = tile >> 8;                // 0..7
  const int col0 = colGroup * 16;
  const int row0 = rowChunk * ROWS_PER_CHUNK;

  // B (4x16 f32) layout: lanes 0-15 hold N=lane, VGPR0=K0, VGPR1=K1;
  //                      lanes 16-31 hold N=lane-16, VGPR0=K2, VGPR1=K3.
  const int nIdx = lane & 15;
  const int kHi = (lane >> 4) * 2;  // 0 or 2
  const float* bPtr = w_in + (size_t)(row0 + kHi) * N_REC + col0 + nIdx;

  v8f c = {};  // 16x16 f32 accumulator, D row M=0 is what we want

  for (int r = 0; r < ROWS_PER_CHUNK; r += 4) {
    // A (16x4 f32): only row M=0 populated. lane0 -> K0,K1 ; lane16 -> K2,K3.
    const float i0 = inputs[row0 + r + 0];
    const float i1 = inputs[row0 + r + 1];
    const float i2 = inputs[row0 + r + 2];
    const float i3 = inputs[row0 + r + 3];
    v2f a;
    a.x = (lane == 0) ? i0 : ((lane == 16) ? i2 : 0.0f);
    a.y = (lane == 0) ? i1 : ((lane == 16) ? i3 : 0.0f);

    v2f b;
    b.x = bPtr[0];      // row (row0+r+kHi)
    b.y = bPtr[N_REC];  // row (row0+r+kHi+1)
    bPtr += (size_t)4 * N_REC;

    // 8-arg form: (neg_a, A, neg_b, B, c_mod, C, reuse_a, reuse_b)
    c = __builtin_amdgcn_wmma_f32_16x16x4_f32(
        false, a, false, b, (short)0, c, false, false);
  }

  // D row M=0 lives in VGPR0 (c[0]) of lanes 0-15, N = lane.
  if (lane < 16) {
    partials[(size_t)rowChunk * N_REC + col0 + lane] = c[0];
  }
}

// ---------------------------------------------------------------------------
// Kernel 2: reduce partials -> isc; LIF dynamics + spike; deterministic
// block-wide sum of spikes -> pop_activity -> surprise. Single block.
// ---------------------------------------------------------------------------
__global__ __launch_bounds__(256) void lif_kernel(
    const float* __restrict__ v_in,
    const float* __restrict__ A_in,
    const float* __restrict__ thresholds,
    const float* __restrict__ partials,
    float* __restrict__ isc_out,
    float* __restrict__ surprise_out,
    float* __restrict__ out_z,
    float* __restrict__ out_v,
    float* __restrict__ out_i,
    float* __restrict__ out_pop) {
  __shared__ float red[256];
  float local = 0.0f;
  const float one_minus_decay = 1.0f - 0.9f;

  for (int j = threadIdx.x; j < N_REC; j += 256) {
    float s = 0.0f;
#pragma unroll
    for (int r = 0; r < ROW_SPLIT; ++r) s += partials[(size_t)r * N_REC + j];
    isc_out[j] = s;

    const float vj = v_in[j];
    float nv = vj + one_minus_decay * (-vj + s);
    const float spk = (tanhf(nv) > thresholds[j]) ? 1.0f : 0.0f;
    nv = nv * (1.0f - spk);
    out_v[j] = nv;
    out_i[j] = nv;       // lif_dynamics2 returns (new_v, new_v, new_z)
    out_z[j] = spk;      // DT == 1.0
    local += spk;
  }

  red[threadIdx.x] = local;
  __syncthreads();
#pragma unroll
  for (int s = 128; s > 0; s >>= 1) {
    if (threadIdx.x < s) red[threadIdx.x] += red[threadIdx.x + s];
    __syncthreads();
  }

  if (threadIdx.x == 0) {
    const float A = A_in[0];
    const float pop = A + (1.0f / 50.0f) * (red[0] - A);
    out_pop[0] = pop;
    const float x = pop / (2.0f * (float)N_REC / (float)N_IN);  // /2
    const float tan = tanhf(x) * ((x > 0.0f) ? 1.0f : 0.0f);
    const float sur =
        0.00157f * tan + 0.05749f * tan * ((x > 0.2206f) ? 1.0f : 0.0f);
    surprise_out[0] = sur;
  }
}

// ---------------------------------------------------------------------------
// Kernel 3: rank-1 update + sign-constraint clip; streaming 64MB r + 64MB w.
// float4 (b128) loads of w_in; b32 stores (d_out+12289 is only 4B aligned).
// Rows k < N_IN/2 are plain copy (mask zeros them, clip bounds are +-inf).
// ---------------------------------------------------------------------------
__global__ __launch_bounds__(256) void wupdate_kernel(
    const float* __restrict__ w_in,
    const float* __restrict__ inputs,
    const float* __restrict__ isc,
    const float* __restrict__ surprise,
    float* __restrict__ new_w) {
  const size_t idx4 = (size_t)blockIdx.x * 256 + threadIdx.x;  // float4 index
  const int k = (int)(idx4 >> 10);        // row (N_REC/4 = 1024 float4 per row)
  const int c4 = (int)(idx4 & 1023);      // float4 index within row
  const int j = c4 * 4;                   // column

  float4 w = ((const float4*)w_in)[idx4];

  if (k >= N_IN / 2) {
    const float coef = -inputs[k] * surprise[0];
    const float4 s = ((const float4*)isc)[c4];
    w.x += coef * s.x;
    w.y += coef * s.y;
    w.z += coef * s.z;
    w.w += coef * s.w;
    if (j < N_REC / 2) {  // clip_max = 0
      w.x = fminf(w.x, 0.0f); w.y = fminf(w.y, 0.0f);
      w.z = fminf(w.z, 0.0f); w.w = fminf(w.w, 0.0f);
    } else {              // clip_min = 0
      w.x = fmaxf(w.x, 0.0f); w.y = fmaxf(w.y, 0.0f);
      w.z = fmaxf(w.z, 0.0f); w.w = fmaxf(w.w, 0.0f);
    }
  }

  const size_t base = idx4 * 4;
  new_w[base + 0] = w.x;
  new_w[base + 1] = w.y;
  new_w[base + 2] = w.z;
  new_w[base + 3] = w.w;
}

// ---------------------------------------------------------------------------
extern "C" void kernel_launch(void* const* d_in, const int* in_sizes, int n_in,
                              void* d_out, int out_size, void* d_ws,
                              size_t ws_size, hipStream_t stream) {
  // setup_inputs order: inputs, v, i_state, z, A, w_in, thresholds
  const float* inputs = (const float*)d_in[0];
  const float* v      = (const float*)d_in[1];
  const float* A      = (const float*)d_in[4];
  const float* w_in   = (const float*)d_in[5];
  const float* thr    = (const float*)d_in[6];

  // outputs concatenated flat: new_z[4096], new_v[4096], new_i[4096],
  // pop_activity[1], new_w_in[4096*4096]
  float* out    = (float*)d_out;
  float* out_z  = out;
  float* out_v  = out + N_REC;
  float* out_i  = out + 2 * N_REC;
  float* out_p  = out + 3 * N_REC;
  float* out_w  = out + 3 * N_REC + 1;

  // scratch layout
  float* partials = (float*)d_ws;                    // ROW_SPLIT * N_REC
  float* isc      = partials + ROW_SPLIT * N_REC;    // N_REC
  float* sur      = isc + N_REC;                     // 1

  const int gemv_blocks = (COL_GROUPS * ROW_SPLIT) / 8;  // 256 blocks, 8 waves each
  gemv_wmma_kernel<<<gemv_blocks, 256, 0, stream>>>(inputs, w_in, partials);

  lif_kernel<<<1, 256, 0, stream>>>(v, A, thr, partials, isc, sur,
                                    out_z, out_v, out_i, out_p);

  const int upd_blocks = (N_IN * N_REC / 4) / 256;  // 16384
  wupdate_kernel<<<upd_blocks, 256, 0, stream>>>(w_in, inputs, isc, sur, out_w);
}